// OptNetf_24902220382769
// MI455X (gfx1250) — compile-verified
//
#include <hip/hip_runtime.h>

// z = min(x, 7.0f) elementwise, 4096x16384 fp32.
// Pure HBM streaming: 536 MB total -> ~23 us floor at 23.3 TB/s.
//
// MI455X shape of the kernel:
//  - wave32, 256-thread blocks (8 waves/WGP-slot)
//  - exactly 4 b128 NT loads in flight per wave before the single wait
//  - immediate-offset addressing: one 64-bit base per thread, IOFFSET for
//    the 4 tiles (4 KB apart) -> near-zero VALU/SALU overhead
//  - NO bounds checks in the hot kernel: grid covers only full tiles,
//    scalar tail kernel (not launched for this shape) covers the rest.

#ifndef BOUND_F
#define BOUND_F 7.0f
#endif

typedef __attribute__((ext_vector_type(4))) float v4f;

#define THREADS 256
#define UNROLL 4

__global__ __launch_bounds__(THREADS) void clamp7_v4_kernel(
    const float* __restrict__ x, float* __restrict__ out) {
  // Each block owns a contiguous span of THREADS*UNROLL float4s.
  long long base =
      (long long)blockIdx.x * (THREADS * UNROLL) + threadIdx.x;
  const v4f* __restrict__ xv = (const v4f*)x + base;
  v4f* __restrict__ ov = (v4f*)out + base;

  v4f v[UNROLL];
  // 4x global_load_b128 th:TH_LOAD_NT, same VADDR, immediate offsets.
#pragma unroll
  for (int u = 0; u < UNROLL; ++u) {
    v[u] = __builtin_nontemporal_load(xv + u * THREADS);
  }

#pragma unroll
  for (int u = 0; u < UNROLL; ++u) {
    v4f r;
    r.x = __builtin_fminf(v[u].x, BOUND_F);
    r.y = __builtin_fminf(v[u].y, BOUND_F);
    r.z = __builtin_fminf(v[u].z, BOUND_F);
    r.w = __builtin_fminf(v[u].w, BOUND_F);
    __builtin_nontemporal_store(r, ov + u * THREADS);
  }
}

// Scalar tail for elements past the last full tile (and any n % 4 != 0).
// Never launched for 4096x16384.
__global__ void clamp7_tail_kernel(const float* __restrict__ x,
                                   float* __restrict__ out,
                                   long long start, long long n) {
  long long i = start + (long long)blockIdx.x * blockDim.x + threadIdx.x;
  if (i < n) {
    out[i] = __builtin_fminf(x[i], BOUND_F);
  }
}

extern "C" void kernel_launch(void* const* d_in, const int* in_sizes, int n_in,
                              void* d_out, int out_size, void* d_ws, size_t ws_size,
                              hipStream_t stream) {
  const float* x = (const float*)d_in[0];
  float* out = (float*)d_out;

  long long n = (long long)in_sizes[0];
  long long n4 = n / 4;                              // full float4s
  long long per_block = (long long)THREADS * UNROLL; // float4s per block
  long long full_blocks = n4 / per_block;

  if (full_blocks > 0) {
    clamp7_v4_kernel<<<(unsigned)full_blocks, THREADS, 0, stream>>>(x, out);
  }

  long long tail_start = full_blocks * per_block * 4; // first element not covered
  long long tail = n - tail_start;
  if (tail > 0) {
    unsigned tblocks = (unsigned)((tail + 255) / 256);
    clamp7_tail_kernel<<<tblocks, 256, 0, stream>>>(x, out, tail_start, n);
  }
}